// GAT_28089086116154
// MI455X (gfx1250) — compile-verified
//
#include <hip/hip_runtime.h>
#include <hip/hip_bf16.h>

typedef __attribute__((ext_vector_type(2))) float v2f;
typedef __attribute__((ext_vector_type(8))) float v8f;

#define NEG_SLOPE 0.2f

// ---------- order-preserving float <-> uint for atomicMax on signed floats ----------
__device__ __forceinline__ unsigned f2ord(float f) {
  unsigned u = __float_as_uint(f);
  return (u & 0x80000000u) ? ~u : (u | 0x80000000u);
}
__device__ __forceinline__ float ord2f(unsigned k) {
  unsigned u = (k & 0x80000000u) ? (k ^ 0x80000000u) : ~k;
  return __uint_as_float(u);
}
__device__ __forceinline__ float lrelu(float v) {
  return v > 0.0f ? v : NEG_SLOPE * v;
}

// =====================================================================
// C[M x 64] = A[M x K] @ B[K x 64] (+ optional bias), full-precision
// f32 WMMA (V_WMMA_F32_16X16X4_F32). One wave -> one 16x16 C tile,
// 4 waves/block cover the 64 output columns. K multiple of 4.
// A layout per §7.12.2: lane L holds A[M=L%16, K = 2*(L/16)+{0,1}]
// B layout:            lane L holds B[K = 2*(L/16)+{0,1}, N=L%16]
// C/D layout: VGPR r -> rows (r, r+8) split by lane halves.
// =====================================================================
__global__ void __launch_bounds__(128)
gemm64_wmma(const float* __restrict__ A, const float* __restrict__ B,
            const float* __restrict__ bias, float* __restrict__ C,
            int M, int K) {
  const int lane  = threadIdx.x & 31;
  const int wave  = threadIdx.x >> 5;      // 0..3 -> 16-column tile
  const int tileM = blockIdx.x << 4;
  const int tileN = wave << 4;
  const int l16   = lane & 15;
  const int khalf = (lane >> 4) << 1;      // 0 or 2

  int arow = tileM + l16;
  if (arow >= M) arow = M - 1;             // clamp (dup read, store guarded)
  const float* arp = A + (long long)arow * K + khalf;
  const float* bcp = B + (long long)khalf * 64 + tileN + l16;

  v8f acc = {};
  for (int k0 = 0; k0 < K; k0 += 4) {
    v2f a, b;
    a.x = arp[k0 + 0];
    a.y = arp[k0 + 1];
    b.x = bcp[(long long)(k0 + 0) * 64];
    b.y = bcp[(long long)(k0 + 1) * 64];
    acc = __builtin_amdgcn_wmma_f32_16x16x4_f32(
        /*neg_a=*/false, a, /*neg_b=*/false, b,
        /*c_mod=*/(short)0, acc, /*reuse_a=*/false, /*reuse_b=*/false);
  }

  const int col  = tileN + l16;
  const float bv = bias ? bias[col] : 0.0f;
  const int rbase = tileM + ((lane >> 4) << 3);
#pragma unroll
  for (int r = 0; r < 8; ++r) {
    const int row = rbase + r;
    if (row < M) C[(long long)row * 64 + col] = acc[r] + bv;
  }
}

// ---------------------------------------------------------------------
// a_s[n,h] = sum_c xw[n,h*16+c] * att_s[h,c]   (and same for a_d)
// one thread per (n,h); C=16 per head
// ---------------------------------------------------------------------
__global__ void attn_coef(const float* __restrict__ xw,
                          const float* __restrict__ ws16,
                          const float* __restrict__ wd16,
                          float* __restrict__ a_s, float* __restrict__ a_d,
                          int Nn) {
  const int i = blockIdx.x * blockDim.x + threadIdx.x;   // i = n*4 + h
  if (i >= Nn * 4) return;
  const int h = i & 3;
  const float* row = xw + (long long)(i >> 2) * 64 + h * 16;
  const float* ws  = ws16 + h * 16;
  const float* wd  = wd16 + h * 16;
  float ss = 0.0f, sd = 0.0f;
#pragma unroll
  for (int c = 0; c < 16; ++c) {
    const float v = row[c];
    ss += v * ws[c];
    sd += v * wd[c];
  }
  a_s[i] = ss;
  a_d[i] = sd;
}

// ---------------------------------------------------------------------
// pass A: per-destination segment max of leaky_relu(a_s[src]+a_d[dst])
// ---------------------------------------------------------------------
__global__ void edge_max(const int* __restrict__ srcI, const int* __restrict__ dstI,
                         const float* __restrict__ a_s, const float* __restrict__ a_d,
                         unsigned* __restrict__ mkey, int E) {
  const int e = blockIdx.x * blockDim.x + threadIdx.x;
  if (e >= E) return;
  const int s = srcI[e], d = dstI[e];
#pragma unroll
  for (int h = 0; h < 4; ++h) {
    const float v = lrelu(a_s[s * 4 + h] + a_d[d * 4 + h]);
    atomicMax(&mkey[d * 4 + h], f2ord(v));
  }
}

// ---------------------------------------------------------------------
// pass B: segment sum of exp(e - max)
// ---------------------------------------------------------------------
__global__ void edge_expsum(const int* __restrict__ srcI, const int* __restrict__ dstI,
                            const float* __restrict__ a_s, const float* __restrict__ a_d,
                            const unsigned* __restrict__ mkey, float* __restrict__ ssum,
                            int E) {
  const int e = blockIdx.x * blockDim.x + threadIdx.x;
  if (e >= E) return;
  const int s = srcI[e], d = dstI[e];
#pragma unroll
  for (int h = 0; h < 4; ++h) {
    const float v = lrelu(a_s[s * 4 + h] + a_d[d * 4 + h]);
    const float m = ord2f(mkey[d * 4 + h]);
    atomicAdd(&ssum[d * 4 + h], __expf(v - m));
  }
}

// ---------------------------------------------------------------------
// pass C: agg[dst] += alpha * xw[src].  16 lanes per edge, each lane
// owns 4 consecutive channels (float4 gather, 4 f32 atomic adds).
// alpha recomputed from (a_s, a_d, m, s) -> no [E,H] buffer.
// ---------------------------------------------------------------------
__global__ void edge_msg(const int* __restrict__ srcI, const int* __restrict__ dstI,
                         const float* __restrict__ a_s, const float* __restrict__ a_d,
                         const unsigned* __restrict__ mkey, const float* __restrict__ ssum,
                         const float* __restrict__ xw, float* __restrict__ agg,
                         long long T) {
  const long long t = (long long)blockIdx.x * blockDim.x + threadIdx.x;
  if (t >= T) return;
  const int e = (int)(t >> 4);
  const int l = (int)(t & 15);      // channel group: channels [4l, 4l+4)
  const int s = srcI[e], d = dstI[e];
  const int h = l >> 2;             // head = (4l)/16

  const float v  = lrelu(a_s[s * 4 + h] + a_d[d * 4 + h]);
  const float m  = ord2f(mkey[d * 4 + h]);
  const float al = __expf(v - m) / (ssum[d * 4 + h] + 1e-16f);

  const float4 xv = *(const float4*)(xw + (long long)s * 64 + l * 4);
  float* o = agg + (long long)d * 64 + l * 4;
  atomicAdd(o + 0, xv.x * al);
  atomicAdd(o + 1, xv.y * al);
  atomicAdd(o + 2, xv.z * al);
  atomicAdd(o + 3, xv.w * al);
}

// ---------------------------------------------------------------------
// h = elu(agg + bias), in place (agg == views[l]); mirror into final_emb
// ---------------------------------------------------------------------
__global__ void bias_elu_copy(float* __restrict__ agg, const float* __restrict__ bias,
                              float* __restrict__ femb /* pre-offset by l*64 */,
                              int Nn) {
  const int i = blockIdx.x * blockDim.x + threadIdx.x;
  if (i >= Nn * 64) return;
  const int c = i & 63;
  const int n = i >> 6;
  float v = agg[i] + bias[c];
  v = v > 0.0f ? v : (__expf(v) - 1.0f);
  agg[i] = v;
  femb[(long long)n * 128 + c] = v;
}

// ---------------------------------------------------------------------
// row softmax over 64 logits; one wave32 per node, 2 values per lane
// ---------------------------------------------------------------------
__global__ void softmax64(const float* __restrict__ logits, float* __restrict__ probs,
                          int Nn) {
  const int n = blockIdx.x * (blockDim.x >> 5) + (threadIdx.x >> 5);
  if (n >= Nn) return;
  const int lane = threadIdx.x & 31;
  const float v0 = logits[(long long)n * 64 + lane];
  const float v1 = logits[(long long)n * 64 + lane + 32];
  float mx = fmaxf(v0, v1);
#pragma unroll
  for (int o = 16; o > 0; o >>= 1) mx = fmaxf(mx, __shfl_xor(mx, o, 32));
  const float e0 = __expf(v0 - mx), e1 = __expf(v1 - mx);
  float sm = e0 + e1;
#pragma unroll
  for (int o = 16; o > 0; o >>= 1) sm += __shfl_xor(sm, o, 32);
  const float inv = 1.0f / sm;
  probs[(long long)n * 64 + lane]      = e0 * inv;
  probs[(long long)n * 64 + lane + 32] = e1 * inv;
}

// =====================================================================
extern "C" void kernel_launch(void* const* d_in, const int* in_sizes, int n_in,
                              void* d_out, int out_size, void* d_ws, size_t ws_size,
                              hipStream_t stream) {
  (void)n_in; (void)out_size; (void)ws_size;
  const float* x   = (const float*)d_in[0];
  const int*   ei  = (const int*)d_in[1];
  const float* W0  = (const float*)d_in[2];
  const float* as0 = (const float*)d_in[3];
  const float* ad0 = (const float*)d_in[4];
  const float* b0  = (const float*)d_in[5];
  const float* W1  = (const float*)d_in[6];
  const float* as1 = (const float*)d_in[7];
  const float* ad1 = (const float*)d_in[8];
  const float* b1  = (const float*)d_in[9];
  const float* lw  = (const float*)d_in[10];
  const float* lb  = (const float*)d_in[11];

  const int N = in_sizes[0] / 128;   // 50000
  const int E = in_sizes[1] / 2;     // 1600000
  const int* srcI = ei;
  const int* dstI = ei + E;

  // ---- output layout: views[2,N,64] | final_emb[N,128] | logits[N,64] | probs[N,64]
  float* out    = (float*)d_out;
  float* views  = out;
  float* femb   = views + (size_t)2 * N * 64;
  float* logits = femb  + (size_t)N * 128;
  float* probs  = logits + (size_t)N * 64;

  // ---- workspace: xw[N,64] | a_s[N,4] | a_d[N,4] | mkey[N,4] | ssum[N,4]  (~16 MB)
  float*    xw   = (float*)d_ws;
  float*    a_s  = xw + (size_t)N * 64;
  float*    a_d  = a_s + (size_t)N * 4;
  unsigned* mkey = (unsigned*)(a_d + (size_t)N * 4);
  float*    ssum = (float*)(mkey + (size_t)N * 4);

  const int gemmGrid = (N + 15) / 16;
  const int eGrid    = (E + 255) / 256;

  for (int l = 0; l < 2; ++l) {
    const float* A   = (l == 0) ? x  : views;   // views[0] == h after layer 0
    const int    K   = (l == 0) ? 128 : 64;
    const float* W   = (l == 0) ? W0  : W1;
    const float* asW = (l == 0) ? as0 : as1;
    const float* adW = (l == 0) ? ad0 : ad1;
    const float* bb  = (l == 0) ? b0  : b1;
    float* agg = views + (size_t)l * N * 64;

    // xw = A @ W  (no bias: GAT adds bias after aggregation)
    gemm64_wmma<<<gemmGrid, 128, 0, stream>>>(A, W, nullptr, xw, N, K);
    attn_coef<<<(N * 4 + 255) / 256, 256, 0, stream>>>(xw, asW, adW, a_s, a_d, N);

    hipMemsetAsync(mkey, 0, (size_t)N * 4 * sizeof(unsigned), stream); // ord(-inf) floor
    hipMemsetAsync(ssum, 0, (size_t)N * 4 * sizeof(float), stream);
    hipMemsetAsync(agg,  0, (size_t)N * 64 * sizeof(float), stream);

    edge_max<<<eGrid, 256, 0, stream>>>(srcI, dstI, a_s, a_d, mkey, E);
    edge_expsum<<<eGrid, 256, 0, stream>>>(srcI, dstI, a_s, a_d, mkey, ssum, E);
    const long long T = (long long)E * 16;
    edge_msg<<<(unsigned)((T + 255) / 256), 256, 0, stream>>>(
        srcI, dstI, a_s, a_d, mkey, ssum, xw, agg, T);

    bias_elu_copy<<<(N * 64 + 255) / 256, 256, 0, stream>>>(
        agg, bb, femb + (size_t)l * 64, N);
  }

  // logits = final_emb @ lw + lb  (K = 128), then row softmax
  gemm64_wmma<<<gemmGrid, 128, 0, stream>>>(femb, lw, lb, logits, N, 128);
  softmax64<<<(N + 7) / 8, 256, 0, stream>>>(logits, probs, N);
}